// WebAttackGenerator_55533927137315
// MI455X (gfx1250) — compile-verified
//
#include <hip/hip_runtime.h>
#include <math.h>

// ---------------------------------------------------------------------------
// WebAttackGenerator: 2-layer LSTM recurrence, T=200 greedy-decode steps.
// fp32 WMMA (v_wmma_f32_16x16x4_f32) for all matmuls: weights (~26MB) are
// L2-resident on MI455X (192MB L2), recurrence is serial, and fp32 keeps the
// argmax/token chain faithful to the fp32 reference.
// ---------------------------------------------------------------------------

typedef __attribute__((ext_vector_type(2))) float v2f;
typedef __attribute__((ext_vector_type(8))) float v8f;

#define B_  128
#define ND_ 128
#define V_  5000
#define E_  256
#define H_  512
#define T_  200
#define G_  (4 * H_)   // 2048 gate outputs per layer

__device__ __forceinline__ float sigf(float x) { return 1.0f / (1.0f + expf(-x)); }

// ---------------------------------------------------------------------------
// Gates matmul: gates[B,4H] = A[B,Ktot] * W^T, where A = [x | h_prev] and
// W = [Wx | Wh] along K. One wave computes one 16x16 tile of gates with
// V_WMMA_F32_16X16X4_F32, stepping K by 4.
//   A frag (16x4 f32): lanes 0-15 hold K=kk,kk+1; lanes 16-31 hold K=kk+2,kk+3
//   B frag (4x16 f32): same per-lane K split; N = lane%16; B[k,n] = W[n,k]
//   C/D (16x16 f32): VGPR v, lanes 0-15 -> M=v, lanes 16-31 -> M=v+8
// LAYER==0: x = concat(embedding[token], noise)  (Kx = 384)
// LAYER==1: x = h0_new                           (Kx = 512)
// ---------------------------------------------------------------------------
template <int LAYER>
__global__ __launch_bounds__(128) void gates_kernel(
    const float* __restrict__ embedding,   // [V,E]       (LAYER 0 only)
    const int*   __restrict__ token,       // [B]         (LAYER 0 only)
    const float* __restrict__ noise,       // [B,ND]      (LAYER 0 only)
    const float* __restrict__ x_h,         // [B,H]       (LAYER 1 only)
    const float* __restrict__ h_prev,      // [B,H]
    const float* __restrict__ Wx,          // [4H,Kx] row-major
    const float* __restrict__ Wh,          // [4H,H]  row-major
    float* __restrict__ gates)             // [B,4H]
{
    constexpr int Kx   = (LAYER == 0) ? (E_ + ND_) : H_;
    constexpr int Ktot = Kx + H_;

    const int lane  = threadIdx.x & 31;
    const int wave  = threadIdx.x >> 5;
    const int tileN = blockIdx.x * 4 + wave;   // 128 N-tiles (4H/16)
    const int tileM = blockIdx.y;              // 8 M-tiles (B/16)
    const int half  = lane >> 4;               // 0: K+0/1, 1: K+2/3
    const int lidx  = lane & 15;
    const int m     = tileM * 16 + lidx;       // activation row this lane feeds
    const int n     = tileN * 16 + lidx;       // weight row (= output column)
    const int kofs  = half * 2;

    const float* emb_row   = nullptr;
    const float* noise_row = nullptr;
    const float* x_row     = nullptr;
    if (LAYER == 0) {
        emb_row   = embedding + (size_t)token[m] * E_;
        noise_row = noise + (size_t)m * ND_;
    } else {
        x_row = x_h + (size_t)m * H_;
    }
    const float* h_row  = h_prev + (size_t)m * H_;
    const float* Wx_row = Wx + (size_t)n * Kx;
    const float* Wh_row = Wh + (size_t)n * H_;

    v8f acc = {};
    for (int kk = 0; kk < Ktot; kk += 4) {
        const int kb = kk + kofs;              // segment bounds are multiples
        v2f a, b;                              // of 4 -> no straddling
        if (kb < Kx) {
            if (LAYER == 0)
                a = (kb < E_) ? *(const v2f*)(emb_row + kb)
                              : *(const v2f*)(noise_row + (kb - E_));
            else
                a = *(const v2f*)(x_row + kb);
            b = *(const v2f*)(Wx_row + kb);
        } else {
            a = *(const v2f*)(h_row + (kb - Kx));
            b = *(const v2f*)(Wh_row + (kb - Kx));
        }
        acc = __builtin_amdgcn_wmma_f32_16x16x4_f32(
            /*neg_a=*/false, a, /*neg_b=*/false, b,
            /*c_mod=*/(short)0, acc, /*reuse_a=*/false, /*reuse_b=*/false);
    }

    const int rowbase = tileM * 16 + half * 8;
    const int col     = tileN * 16 + lidx;
#pragma unroll
    for (int v = 0; v < 8; ++v)
        gates[(size_t)(rowbase + v) * G_ + col] = acc[v];
}

// ---------------------------------------------------------------------------
// Elementwise LSTM cell update (PyTorch gate order i,f,g,o), biases folded in.
// ---------------------------------------------------------------------------
__global__ __launch_bounds__(256) void lstm_update_kernel(
    const float* __restrict__ gates,  // [B,4H]
    const float* __restrict__ b_ih,   // [4H]
    const float* __restrict__ b_hh,   // [4H]
    float* __restrict__ h,            // [B,H] in/out
    float* __restrict__ c)            // [B,H] in/out
{
    const int idx = blockIdx.x * blockDim.x + threadIdx.x;  // B*H threads
    const int b = idx / H_;
    const int n = idx % H_;
    const size_t g0 = (size_t)b * G_;
    const float gi = gates[g0 + n]          + b_ih[n]          + b_hh[n];
    const float gf = gates[g0 + H_ + n]     + b_ih[H_ + n]     + b_hh[H_ + n];
    const float gg = gates[g0 + 2 * H_ + n] + b_ih[2 * H_ + n] + b_hh[2 * H_ + n];
    const float go = gates[g0 + 3 * H_ + n] + b_ih[3 * H_ + n] + b_hh[3 * H_ + n];
    const float i = sigf(gi), f = sigf(gf), o = sigf(go), g = tanhf(gg);
    const float cn = f * c[idx] + i * g;
    c[idx] = cn;
    h[idx] = o * tanhf(cn);
}

// ---------------------------------------------------------------------------
// Logits matmul: out[b, t, v] = h1[b,:] . W_out[v,:] + b_out[v], fp32 WMMA.
// ---------------------------------------------------------------------------
__global__ __launch_bounds__(128) void logits_kernel(
    const float* __restrict__ h1,     // [B,H]
    const float* __restrict__ Wout,   // [V,H]
    const float* __restrict__ bout,   // [V]
    float* __restrict__ out,          // [B,T,V]
    const int t)
{
    const int lane  = threadIdx.x & 31;
    const int wave  = threadIdx.x >> 5;
    const int tileN = blockIdx.x * 4 + wave;
    if (tileN * 16 >= V_) return;              // wave-uniform tail guard
    const int tileM = blockIdx.y;
    const int half  = lane >> 4;
    const int lidx  = lane & 15;
    const int m     = tileM * 16 + lidx;
    const int n     = tileN * 16 + lidx;
    const int nc    = (n < V_) ? n : (V_ - 1); // clamp OOB weight rows
    const int kofs  = half * 2;

    const float* a_row = h1 + (size_t)m * H_;
    const float* w_row = Wout + (size_t)nc * H_;

    v8f acc = {};
    for (int kk = 0; kk < H_; kk += 4) {
        const int kb = kk + kofs;
        v2f a = *(const v2f*)(a_row + kb);
        v2f b = *(const v2f*)(w_row + kb);
        acc = __builtin_amdgcn_wmma_f32_16x16x4_f32(
            false, a, false, b, (short)0, acc, false, false);
    }

    const int rowbase = tileM * 16 + half * 8;
    const int col     = tileN * 16 + lidx;
    if (col < V_) {
        const float bo = bout[col];
#pragma unroll
        for (int v = 0; v < 8; ++v)
            out[(size_t)(rowbase + v) * ((size_t)T_ * V_) + (size_t)t * V_ + col] =
                acc[v] + bo;
    }
}

// ---------------------------------------------------------------------------
// Greedy argmax over V=5000 logits per batch row (first-max-index on ties,
// matching jnp.argmax). One block per row.
// ---------------------------------------------------------------------------
__global__ __launch_bounds__(256) void argmax_kernel(
    const float* __restrict__ out,    // [B,T,V]
    const int t,
    int* __restrict__ token)          // [B]
{
    const int b   = blockIdx.x;
    const int tid = threadIdx.x;
    const float* row = out + (size_t)b * ((size_t)T_ * V_) + (size_t)t * V_;

    float best = -INFINITY;
    int   bi   = 0x7FFFFFFF;
    for (int v = tid; v < V_; v += 256) {
        const float x = row[v];
        if (x > best || (x == best && v < bi)) { best = x; bi = v; }
    }
    __shared__ float sv[256];
    __shared__ int   si[256];
    sv[tid] = best; si[tid] = bi;
    __syncthreads();
    for (int s = 128; s > 0; s >>= 1) {
        if (tid < s) {
            const float ov = sv[tid + s];
            const int   oi = si[tid + s];
            if (ov > sv[tid] || (ov == sv[tid] && oi < si[tid])) {
                sv[tid] = ov; si[tid] = oi;
            }
        }
        __syncthreads();
    }
    if (tid == 0) token[b] = si[0];
}

// ---------------------------------------------------------------------------
// Tiny heads on final h1: attack logits [B,8] and sophistication [B,16].
// ---------------------------------------------------------------------------
__global__ __launch_bounds__(32) void heads_kernel(
    const float* __restrict__ h1,     // [B,H]
    const float* __restrict__ Watt,   // [8,H]
    const float* __restrict__ batt,   // [8]
    const float* __restrict__ Wsoph,  // [16,H]
    const float* __restrict__ bsoph,  // [16]
    float* __restrict__ out)          // d_out base
{
    const int b = blockIdx.x;
    const int j = threadIdx.x;
    const float* hrow = h1 + (size_t)b * H_;
    const size_t att_base  = (size_t)B_ * T_ * V_;
    const size_t soph_base = att_base + (size_t)B_ * 8;
    if (j < 8) {
        const float* w = Watt + (size_t)j * H_;
        float acc = batt[j];
        for (int k = 0; k < H_; ++k) acc += hrow[k] * w[k];
        out[att_base + (size_t)b * 8 + j] = acc;
    } else if (j < 24) {
        const int jj = j - 8;
        const float* w = Wsoph + (size_t)jj * H_;
        float acc = bsoph[jj];
        for (int k = 0; k < H_; ++k) acc += hrow[k] * w[k];
        out[soph_base + (size_t)b * 16 + jj] = acc;
    }
}

// ---------------------------------------------------------------------------
// Init: zero h0/c0/h1/c1 and set start token = 1.
// ---------------------------------------------------------------------------
__global__ __launch_bounds__(256) void init_kernel(float* __restrict__ state,
                                                   int* __restrict__ token)
{
    const int i = blockIdx.x * 256 + threadIdx.x;
    if (i < 4 * B_ * H_) state[i] = 0.0f;
    if (i < B_) token[i] = 1;
}

// ---------------------------------------------------------------------------
// Host-side launch. Workspace layout (floats): h0, c0, h1, c1 (B*H each),
// gates (B*4H), then token[B] as ints.  Total ~2.0 MB + 512 B.
// ---------------------------------------------------------------------------
extern "C" void kernel_launch(void* const* d_in, const int* in_sizes, int n_in,
                              void* d_out, int out_size, void* d_ws, size_t ws_size,
                              hipStream_t stream) {
    const float* noise     = (const float*)d_in[0];
    const float* embedding = (const float*)d_in[1];
    const float* W_ih0     = (const float*)d_in[2];
    const float* W_hh0     = (const float*)d_in[3];
    const float* b_ih0     = (const float*)d_in[4];
    const float* b_hh0     = (const float*)d_in[5];
    const float* W_ih1     = (const float*)d_in[6];
    const float* W_hh1     = (const float*)d_in[7];
    const float* b_ih1     = (const float*)d_in[8];
    const float* b_hh1     = (const float*)d_in[9];
    const float* W_out     = (const float*)d_in[10];
    const float* b_out     = (const float*)d_in[11];
    const float* W_att     = (const float*)d_in[12];
    const float* b_att     = (const float*)d_in[13];
    const float* W_soph    = (const float*)d_in[14];
    const float* b_soph    = (const float*)d_in[15];
    (void)in_sizes; (void)n_in; (void)out_size; (void)ws_size;

    float* ws    = (float*)d_ws;
    float* h0    = ws;
    float* c0    = ws + 1 * B_ * H_;
    float* h1    = ws + 2 * B_ * H_;
    float* c1    = ws + 3 * B_ * H_;
    float* gates = ws + 4 * B_ * H_;              // B*4H floats
    int*   token = (int*)(ws + 4 * B_ * H_ + (size_t)B_ * G_);
    float* out   = (float*)d_out;

    init_kernel<<<(4 * B_ * H_ + 255) / 256, 256, 0, stream>>>(ws, token);

    const dim3 gatesGrid(32, 8);                  // 128 N-tiles x 8 M-tiles, 4 waves/blk
    const dim3 logitsGrid(79, 8);                 // ceil(313 N-tiles / 4) x 8 M-tiles

    for (int t = 0; t < T_; ++t) {
        gates_kernel<0><<<gatesGrid, 128, 0, stream>>>(
            embedding, token, noise, nullptr, h0, W_ih0, W_hh0, gates);
        lstm_update_kernel<<<(B_ * H_) / 256, 256, 0, stream>>>(
            gates, b_ih0, b_hh0, h0, c0);
        gates_kernel<1><<<gatesGrid, 128, 0, stream>>>(
            nullptr, nullptr, nullptr, h0, h1, W_ih1, W_hh1, gates);
        lstm_update_kernel<<<(B_ * H_) / 256, 256, 0, stream>>>(
            gates, b_ih1, b_hh1, h1, c1);
        logits_kernel<<<logitsGrid, 128, 0, stream>>>(h1, W_out, b_out, out, t);
        argmax_kernel<<<B_, 256, 0, stream>>>(out, t, token);
    }

    heads_kernel<<<B_, 32, 0, stream>>>(h1, W_att, b_att, W_soph, b_soph, out);
}